// GATConv_22024592294250
// MI455X (gfx1250) — compile-verified
//
#include <hip/hip_runtime.h>
#include <hip/hip_bf16.h>

typedef __attribute__((ext_vector_type(2))) float v2f;
typedef __attribute__((ext_vector_type(8))) float v8f;

#define IN_DIM   256
#define E_DIM    64
#define N_HEADS  8
#define OUT_DIM  32
#define HD       256     // N_HEADS * OUT_DIM
#define NEG_SLOPE 0.2f

// ---------------------------------------------------------------------------
// Async global -> LDS 16-byte transfer (gfx1250 GLOBAL_LOAD_ASYNC_TO_LDS_B128,
// GV addressing mode: LDS offset in the VDST VGPR, global address in VADDR).
// LDS offsets are the low 32 bits of the generic shared-aperture pointer.
__device__ __forceinline__ void async_copy_b128(void* lds_dst, const void* gsrc) {
    unsigned lds_off = (unsigned)(uintptr_t)lds_dst;
    asm volatile("global_load_async_to_lds_b128 %0, %1, off"
                 :: "v"(lds_off), "v"(gsrc)
                 : "memory");
}
__device__ __forceinline__ void async_wait_all() {
    asm volatile("s_wait_asynccnt 0x0" ::: "memory");
}

// ---------------------------------------------------------------------------
// float atomic max via signed/unsigned integer monotonicity trick.
// Storage must be initialized to -inf (0xFF800000).
__device__ __forceinline__ void atomicMaxFloat(float* addr, float val) {
    if (val >= 0.0f) {
        atomicMax((int*)addr, __float_as_int(val));
    } else {
        atomicMin((unsigned int*)addr, (unsigned int)__float_as_int(val));
    }
}

// ---------------------------------------------------------------------------
// Init per-(node,head) buffers: emax = -inf, denom = 0
__global__ __launch_bounds__(256) void init_nh_kernel(float* __restrict__ emax,
                                                      float* __restrict__ denom,
                                                      int n) {
    int i = blockIdx.x * 256 + threadIdx.x;
    if (i < n) {
        emax[i]  = __int_as_float(0xFF800000);  // -inf
        denom[i] = 0.0f;
    }
}

// Init output with broadcast bias (bias is [H,D] = 256 floats flat)
__global__ __launch_bounds__(256) void init_out_kernel(float* __restrict__ out,
                                                       const float* __restrict__ bias,
                                                       int total) {
    int i = blockIdx.x * 256 + threadIdx.x;
    if (i < total) out[i] = bias[i & (HD - 1)];
}

// ---------------------------------------------------------------------------
// Fold W_e [64,256] with attn_e [8,32] -> Ve [64,16] (cols 8..15 zero-padded)
//   Ve[k][h] = sum_d W_e[k, h*32+d] * attn_e[h, d]
__global__ __launch_bounds__(256) void fold_ve_kernel(const float* __restrict__ W_e,
                                                      const float* __restrict__ attn_e,
                                                      float* __restrict__ Ve) {
    int i = blockIdx.x * 256 + threadIdx.x;   // 1024 entries
    if (i >= E_DIM * 16) return;
    int k = i >> 4;
    int h = i & 15;
    float v = 0.0f;
    if (h < N_HEADS) {
        const float* wr = W_e + (size_t)k * HD + h * OUT_DIM;
        const float* ar = attn_e + h * OUT_DIM;
        #pragma unroll
        for (int d = 0; d < OUT_DIM; ++d) v += wr[d] * ar[d];
    }
    Ve[i] = v;
}

// ---------------------------------------------------------------------------
// GEMM 1: ft = feat[N,256] @ W_fc[256,256], fused eh/et epilogue.
// Block = 256 threads (8 waves). One 16-row M tile per block.
// Wave w owns columns [w*32, w*32+32) == head w exactly (2 WMMA C-tiles).
// A tile staged straight into LDS via async global->LDS (ASYNCcnt path);
// padded stride 260 keeps the ds_load_b64 fragment reads bank-conflict-free.
//
// V_WMMA_F32_16X16X4_F32 fragment model (ISA 7.12.2, 32-bit A 16x4):
//   A: lanes 0-15 rows 0-15 hold K={k0,k0+1}; lanes 16-31 hold K={k0+2,k0+3}
//   B: lane n (0-15) holds B[k0][n],B[k0+1][n]; lanes 16-31 hold K=k0+2,k0+3
//   C/D: VGPR r, lanes 0-15 -> (M=r, N=lane); lanes 16-31 -> (M=r+8, N=lane-16)
__global__ __launch_bounds__(256) void gemm_ft_kernel(const float* __restrict__ feat,
                                                      const float* __restrict__ W_fc,
                                                      const float* __restrict__ attn_h,
                                                      const float* __restrict__ attn_t,
                                                      float* __restrict__ ft,
                                                      float* __restrict__ eh,
                                                      float* __restrict__ et,
                                                      int n_nodes) {
    __shared__ float As[16 * 260];             // 16 rows, padded stride (1040 B, 16B-mult)
    const int tile = blockIdx.x;
    const int row0 = tile * 16;
    const int tid  = threadIdx.x;

    // Async-stage A tile (16 x 256 floats) as 16B packets, memory -> LDS direct.
    for (int i = tid; i < 16 * 64; i += 256) {
        int r  = i >> 6;
        int c4 = i & 63;
        int gr = row0 + r;
        if (gr >= n_nodes) gr = n_nodes - 1;          // clamp (grid is exact anyway)
        async_copy_b128(&As[r * 260 + c4 * 4],
                        feat + (size_t)gr * IN_DIM + c4 * 4);
    }
    async_wait_all();
    __syncthreads();

    const int wave  = tid >> 5;        // == head index
    const int lane  = tid & 31;
    const int lhalf = lane >> 4;       // 0 or 1
    const int lm    = lane & 15;
    const int ncol0 = wave * 32 + lm;  // B column for tile 0

    v8f c0 = {};
    v8f c1 = {};

    #pragma unroll 4
    for (int k0 = 0; k0 < IN_DIM; k0 += 4) {
        const int ka = k0 + lhalf * 2;
        v2f a;
        a.x = As[lm * 260 + ka];
        a.y = As[lm * 260 + ka + 1];

        const float* Wb = W_fc + (size_t)ka * HD;
        v2f b0, b1;
        b0.x = Wb[ncol0];            b0.y = Wb[HD + ncol0];
        b1.x = Wb[ncol0 + 16];       b1.y = Wb[HD + ncol0 + 16];

        c0 = __builtin_amdgcn_wmma_f32_16x16x4_f32(false, a, false, b0, (short)0, c0, false, false);
        c1 = __builtin_amdgcn_wmma_f32_16x16x4_f32(false, a, false, b1, (short)0, c1, false, false);
    }

    // Epilogue: store ft slice; fold per-head eh/et via half-wave reduction.
    const float ah0 = attn_h[wave * OUT_DIM + lm];
    const float ah1 = attn_h[wave * OUT_DIM + lm + 16];
    const float at0 = attn_t[wave * OUT_DIM + lm];
    const float at1 = attn_t[wave * OUT_DIM + lm + 16];

    #pragma unroll
    for (int r = 0; r < 8; ++r) {
        const int m    = r + lhalf * 8;
        const int grow = row0 + m;
        const float v0 = c0[r];
        const float v1 = c1[r];
        if (grow < n_nodes) {
            float* fp = ft + (size_t)grow * HD + wave * 32 + lm;
            fp[0]  = v0;
            fp[16] = v1;
        }
        float ph = v0 * ah0 + v1 * ah1;
        float pt = v0 * at0 + v1 * at1;
        // reduce over the 16 lanes of this half-wave (xor masks stay in-half)
        #pragma unroll
        for (int off = 1; off < 16; off <<= 1) {
            ph += __shfl_xor(ph, off, 32);
            pt += __shfl_xor(pt, off, 32);
        }
        if (lm == 0 && grow < n_nodes) {
            eh[(size_t)grow * N_HEADS + wave] = ph;
            et[(size_t)grow * N_HEADS + wave] = pt;
        }
    }
}

// ---------------------------------------------------------------------------
// GEMM 2 (folded) + logits: ee = edge_attr[E,64] @ Ve[64,16-padded],
// fused per-edge logit + leaky_relu + atomic segment max.
// Block = 8 waves; each wave computes one 16-edge tile (16x16 WMMA, cols 0..7 valid).
__global__ __launch_bounds__(256) void edge_logit_kernel(const float* __restrict__ edge_attr,
                                                         const float* __restrict__ Ve,
                                                         const int* __restrict__ src,
                                                         const int* __restrict__ dst,
                                                         const float* __restrict__ eh,
                                                         const float* __restrict__ et,
                                                         float* __restrict__ elog,
                                                         float* __restrict__ emax,
                                                         int n_edges) {
    __shared__ float Bs[E_DIM * 16];           // 1024 floats
    const int tid = threadIdx.x;
    // Async-stage Ve (256 x 16B packets, one per thread), memory -> LDS direct.
    async_copy_b128(&Bs[tid * 4], Ve + tid * 4);
    async_wait_all();
    __syncthreads();

    const int wave  = tid >> 5;
    const int lane  = tid & 31;
    const int lhalf = lane >> 4;
    const int lm    = lane & 15;
    const int tile  = blockIdx.x * 8 + wave;
    const int row0  = tile * 16;
    if (row0 >= n_edges) return;   // wave-uniform

    v8f c = {};
    #pragma unroll
    for (int k0 = 0; k0 < E_DIM; k0 += 4) {
        const int ka = k0 + lhalf * 2;
        const float* ar = edge_attr + (size_t)(row0 + lm) * E_DIM + ka;
        v2f a;
        a.x = ar[0];
        a.y = ar[1];
        v2f b;
        b.x = Bs[ka * 16 + lm];
        b.y = Bs[(ka + 1) * 16 + lm];
        c = __builtin_amdgcn_wmma_f32_16x16x4_f32(false, a, false, b, (short)0, c, false, false);
    }

    // lane lm holds column h = lm for rows r + 8*lhalf; heads live in cols 0..7
    if (lm < N_HEADS) {
        const int h = lm;
        #pragma unroll
        for (int r = 0; r < 8; ++r) {
            const int eidx = row0 + r + lhalf * 8;
            const int s = src[eidx];
            const int d = dst[eidx];
            float v = eh[(size_t)s * N_HEADS + h] + c[r] + et[(size_t)d * N_HEADS + h];
            v = (v > 0.0f) ? v : v * NEG_SLOPE;          // leaky_relu
            elog[(size_t)eidx * N_HEADS + h] = v;
            atomicMaxFloat(&emax[(size_t)d * N_HEADS + h], v);
        }
    }
}

// ---------------------------------------------------------------------------
// ex = exp(e - emax[dst]); denom[dst] += ex (segment sum). One thread per (edge,head).
__global__ __launch_bounds__(256) void exp_denom_kernel(const int* __restrict__ dst,
                                                        const float* __restrict__ emax,
                                                        float* __restrict__ elog,
                                                        float* __restrict__ denom,
                                                        int total) {
    int i = blockIdx.x * 256 + threadIdx.x;
    if (i >= total) return;
    const int e = i >> 3;
    const int h = i & 7;
    const int d = dst[e];
    const float m = emax[(size_t)d * N_HEADS + h];   // finite: d appears as a dst
    const float x = __expf(elog[i] - m);
    elog[i] = x;                                     // overwrite with ex
    atomicAdd(&denom[(size_t)d * N_HEADS + h], x);
}

// ---------------------------------------------------------------------------
// Weighted scatter aggregation: out[dst] += ft[src] * (ex/denom[dst]).
// One block (256 threads) per edge; warp w handles head w -> the coefficient
// load is warp-uniform (broadcast), ft gathers hit L2 (ft = 51 MB < 192 MB L2).
__global__ __launch_bounds__(256) void aggregate_kernel(const float* __restrict__ ft,
                                                        const float* __restrict__ ex,
                                                        const float* __restrict__ denom,
                                                        const int* __restrict__ src,
                                                        const int* __restrict__ dst,
                                                        float* __restrict__ out,
                                                        int n_edges) {
    const int e = blockIdx.x;
    const int t = threadIdx.x;
    const int h = t >> 5;                      // warp-uniform head
    const int s = src[e];
    const int d = dst[e];
    const float a = ex[(size_t)e * N_HEADS + h] / denom[(size_t)d * N_HEADS + h];
    const float* fp = ft + (size_t)s * HD;
    __builtin_prefetch(fp + t, 0, 3);          // global_prefetch_b8
    const float val = fp[t] * a;
    atomicAdd(&out[(size_t)d * HD + t], val);
}

// ---------------------------------------------------------------------------
extern "C" void kernel_launch(void* const* d_in, const int* in_sizes, int n_in,
                              void* d_out, int out_size, void* d_ws, size_t ws_size,
                              hipStream_t stream) {
    const float* feat      = (const float*)d_in[0];
    const float* edge_attr = (const float*)d_in[1];
    const int*   src       = (const int*)  d_in[2];
    const int*   dst       = (const int*)  d_in[3];
    const float* W_fc      = (const float*)d_in[4];
    const float* W_e       = (const float*)d_in[5];
    const float* attn_h    = (const float*)d_in[6];
    const float* attn_e    = (const float*)d_in[7];
    const float* attn_t    = (const float*)d_in[8];
    const float* bias      = (const float*)d_in[9];
    float*       out       = (float*)d_out;

    const int n_nodes = in_sizes[0] / IN_DIM;
    const int n_edges = in_sizes[2];
    const int nh      = n_nodes * N_HEADS;

    // Workspace carve-out (256B aligned slices)
    char* ws = (char*)d_ws;
    size_t off = 0;
    auto carve = [&](size_t floats) {
        float* p = (float*)(ws + off);
        off += ((floats * sizeof(float)) + 255) & ~(size_t)255;
        return p;
    };
    float* ft    = carve((size_t)n_nodes * HD);       // projected node features
    float* eh    = carve(nh);                         // source-side scores
    float* et    = carve(nh);                         // target-side scores
    float* emax  = carve(nh);                         // segment max
    float* denom = carve(nh);                         // segment sum of exp
    float* elog  = carve((size_t)n_edges * N_HEADS);  // logits, then ex
    float* Ve    = carve(E_DIM * 16);                 // folded W_e @ attn_e (padded)

    // 1) init emax/denom and bias-broadcast output
    init_nh_kernel<<<(nh + 255) / 256, 256, 0, stream>>>(emax, denom, nh);
    init_out_kernel<<<((size_t)n_nodes * HD + 255) / 256, 256, 0, stream>>>(
        out, bias, n_nodes * HD);

    // 2) fold W_e with attn_e -> Ve[64,16]
    fold_ve_kernel<<<(E_DIM * 16 + 255) / 256, 256, 0, stream>>>(W_e, attn_e, Ve);

    // 3) node projection GEMM (WMMA) + fused eh/et
    gemm_ft_kernel<<<(n_nodes + 15) / 16, 256, 0, stream>>>(
        feat, W_fc, attn_h, attn_t, ft, eh, et, n_nodes);

    // 4) edge GEMM (WMMA) + logits + atomic segment max
    edge_logit_kernel<<<(n_edges + 127) / 128, 256, 0, stream>>>(
        edge_attr, Ve, src, dst, eh, et, elog, emax, n_edges);

    // 5) exp + segment-sum denominators
    exp_denom_kernel<<<((size_t)n_edges * N_HEADS + 255) / 256, 256, 0, stream>>>(
        dst, emax, elog, denom, n_edges * N_HEADS);

    // 6) weighted scatter aggregation
    aggregate_kernel<<<n_edges, 256, 0, stream>>>(ft, elog, denom, src, dst, out, n_edges);
}